// UniversalMotiveEngine_36833639530798
// MI455X (gfx1250) — compile-verified
//
#include <hip/hip_runtime.h>
#include <hip/hip_bf16.h>

typedef float v2f __attribute__((ext_vector_type(2)));
typedef float v8f __attribute__((ext_vector_type(8)));

#define NROW 8192
#define DIMA 512
#define HIDD 64
#define MOT  64
#define NM   (NROW * MOT)

// D = A(16x4) * B(4x16) + C, full fp32 on the WMMA unit.
static __device__ __forceinline__ v8f wmma_f32(v2f a, v2f b, v8f c) {
  return __builtin_amdgcn_wmma_f32_16x16x4_f32(false, a, false, b, (short)0, c,
                                               false, false);
}

// ---------------------------------------------------------------------------
// Stage 1: M = relu(X @ W1 + b1) @ W2 + b2   (per wave: 16 rows, full width)
// blockIdx.x in [0,64): 128-row block; blockIdx.y in {0,1}: set A / set B
// ---------------------------------------------------------------------------
__global__ __launch_bounds__(256) void functor_kernel(
    const float* __restrict__ XA, const float* __restrict__ XB,
    const float* __restrict__ WA1, const float* __restrict__ bA1v,
    const float* __restrict__ WA2, const float* __restrict__ bA2v,
    const float* __restrict__ WB1, const float* __restrict__ bB1v,
    const float* __restrict__ WB2, const float* __restrict__ bB2v,
    float* __restrict__ out) {
  __shared__ float Hl[8 * 16 * 64];  // 32 KB: one 16x64 H tile per wave
  const int tid = threadIdx.x;
  const int w = tid >> 5, lane = tid & 31;
  const int lo = lane & 15, hi = lane >> 4;
  const int set = blockIdx.y;
  const float* X  = set ? XB  : XA;
  const float* W1 = set ? WB1 : WA1;
  const float* B1 = set ? bB1v : bA1v;
  const float* W2 = set ? WB2 : WA2;
  const float* B2 = set ? bB2v : bA2v;
  float* M = out + 1 + (size_t)set * NM;
  const int i0 = blockIdx.x * 128 + w * 16;

  // GEMM1: H(16x64) = X(16x512) @ W1(512x64), K-steps of 4
  v8f accH[4] = {};
  const float* xp  = X + (size_t)(i0 + lo) * DIMA + 2 * hi;   // A frag: b64
  const float* w1p = W1 + (size_t)(2 * hi) * HIDD + lo;       // B frag: 2x b32
#pragma unroll 4
  for (int t = 0; t < DIMA / 4; ++t) {
    v2f a = *(const v2f*)(xp + 4 * t);
#pragma unroll
    for (int jt = 0; jt < 4; ++jt) {
      v2f b;
      b.x = w1p[(size_t)(4 * t) * HIDD + 16 * jt];
      b.y = w1p[(size_t)(4 * t + 1) * HIDD + 16 * jt];
      accH[jt] = wmma_f32(a, b, accH[jt]);
    }
  }
  // bias + relu, spill C-layout tile to LDS so we can re-read in A-layout
  float* hb = Hl + w * (16 * 64);
#pragma unroll
  for (int jt = 0; jt < 4; ++jt) {
    float bias = B1[lo + 16 * jt];
#pragma unroll
    for (int r = 0; r < 8; ++r) {
      float v = accH[jt][r] + bias;
      hb[(r + 8 * hi) * 64 + lo + 16 * jt] = v > 0.f ? v : 0.f;
    }
  }
  __syncthreads();

  // GEMM2: M(16x64) = H(16x64) @ W2(64x64)
  v8f accM[4] = {};
  const float* hp  = hb + lo * 64 + 2 * hi;                   // ds_load_b64
  const float* w2p = W2 + (2 * hi) * MOT + lo;
#pragma unroll
  for (int t = 0; t < HIDD / 4; ++t) {
    v2f a = *(const v2f*)(hp + 4 * t);
#pragma unroll
    for (int jt = 0; jt < 4; ++jt) {
      v2f b;
      b.x = w2p[(4 * t) * MOT + 16 * jt];
      b.y = w2p[(4 * t + 1) * MOT + 16 * jt];
      accM[jt] = wmma_f32(a, b, accM[jt]);
    }
  }
#pragma unroll
  for (int jt = 0; jt < 4; ++jt) {
    float bias = B2[lo + 16 * jt];
#pragma unroll
    for (int r = 0; r < 8; ++r)
      M[(size_t)(i0 + r + 8 * hi) * MOT + lo + 16 * jt] = accM[jt][r] + bias;
  }
}

// ---------------------------------------------------------------------------
// Stage 2a: per-row squared norms (M_A rows then M_B rows, contiguous)
// ---------------------------------------------------------------------------
__global__ __launch_bounds__(256) void norms_kernel(const float* __restrict__ out,
                                                    float* __restrict__ nrm) {
  int idx = blockIdx.x * blockDim.x + threadIdx.x;
  if (idx >= 2 * NROW) return;
  const float* m = out + 1 + (size_t)idx * MOT;
  float s = 0.f;
#pragma unroll
  for (int c = 0; c < MOT; ++c) { float v = m[c]; s += v * v; }
  nrm[idx] = s;
}

// ---------------------------------------------------------------------------
// Stage 2b: one 128x128 tile of one kernel matrix per block.
// z=0: (A,A)  z=1: (B,B)  z=2: (A,B).  Deterministic per-block double partial.
// ---------------------------------------------------------------------------
__global__ __launch_bounds__(256) void mmd_tile_kernel(
    const float* __restrict__ out, const float* __restrict__ nrm,
    double* __restrict__ partial) {
  __shared__ float sX[128 * 64];
  __shared__ float sY[128 * 64];
  __shared__ double wsum[8];
  const int tid = threadIdx.x;
  const int w = tid >> 5, lane = tid & 31;
  const int lo = lane & 15, hi = lane >> 4;
  const int bx = blockIdx.x, by = blockIdx.y, z = blockIdx.z;
  const float* MA = out + 1;
  const float* MB = out + 1 + NM;
  const float* PX = (z == 1) ? MB : MA;
  const float* PY = (z == 0) ? MA : MB;
  const float* nX = nrm + ((z == 1) ? NROW : 0);
  const float* nY = nrm + ((z == 0) ? 0 : NROW);

  for (int i = tid; i < 128 * 64; i += 256) {   // stage tiles into LDS (aligned)
    sX[i] = PX[(size_t)by * (128 * 64) + i];
    sY[i] = PY[(size_t)bx * (128 * 64) + i];
  }
  __syncthreads();

  float nx[8];
#pragma unroll
  for (int r = 0; r < 8; ++r) nx[r] = nX[by * 128 + 16 * w + 8 * hi + r];

  float lsum = 0.f;
  const float* ap = sX + (16 * w + lo) * 64 + 2 * hi;
  for (int jj = 0; jj < 8; ++jj) {
    v8f acc = {};
    const float* bp = sY + (16 * jj + lo) * 64 + 2 * hi;
#pragma unroll
    for (int t = 0; t < 16; ++t) {
      v2f a = *(const v2f*)(ap + 4 * t);   // A: rows of X-tile
      v2f b = *(const v2f*)(bp + 4 * t);   // B = Y-tile^T: consecutive-K b64
      acc = wmma_f32(a, b, acc);
    }
    float ny = nY[bx * 128 + 16 * jj + lo];
#pragma unroll
    for (int r = 0; r < 8; ++r) {
      float d = nx[r] + ny - 2.f * acc[r];
      d = d > 0.f ? d : 0.f;               // clamp like reference
      lsum += __expf(-0.5f * d);           // sigma = 1
    }
  }
#pragma unroll
  for (int off = 16; off > 0; off >>= 1) lsum += __shfl_xor(lsum, off, 32);
  if (lane == 0) wsum[w] = (double)lsum;
  __syncthreads();
  if (tid == 0) {
    double s = 0.0;
    for (int i = 0; i < 8; ++i) s += wsum[i];
    partial[(size_t)z * 4096 + by * 64 + bx] = s;
  }
}

// ---------------------------------------------------------------------------
// Stage 3: reduce 3 x 4096 double partials -> mmd loss in d_out[0]
// ---------------------------------------------------------------------------
__global__ __launch_bounds__(256) void mmd_final_kernel(
    const double* __restrict__ partial, float* __restrict__ out) {
  __shared__ double red[3][256];
  int tid = threadIdx.x;
  double s0 = 0, s1 = 0, s2 = 0;
  for (int i = tid; i < 4096; i += 256) {
    s0 += partial[i];
    s1 += partial[4096 + i];
    s2 += partial[8192 + i];
  }
  red[0][tid] = s0; red[1][tid] = s1; red[2][tid] = s2;
  __syncthreads();
  for (int off = 128; off > 0; off >>= 1) {
    if (tid < off) {
      red[0][tid] += red[0][tid + off];
      red[1][tid] += red[1][tid + off];
      red[2][tid] += red[2][tid + off];
    }
    __syncthreads();
  }
  if (tid == 0) {
    double inv = 1.0 / ((double)NROW * (double)NROW);
    out[0] = (float)((red[0][0] + red[1][0] - 2.0 * red[2][0]) * inv);
  }
}

extern "C" void kernel_launch(void* const* d_in, const int* in_sizes, int n_in,
                              void* d_out, int out_size, void* d_ws, size_t ws_size,
                              hipStream_t stream) {
  const float* XA  = (const float*)d_in[0];
  const float* XB  = (const float*)d_in[1];
  const float* WA1 = (const float*)d_in[2];
  const float* bA1 = (const float*)d_in[3];
  const float* WA2 = (const float*)d_in[4];
  const float* bA2 = (const float*)d_in[5];
  const float* WB1 = (const float*)d_in[6];
  const float* bB1 = (const float*)d_in[7];
  const float* WB2 = (const float*)d_in[8];
  const float* bB2 = (const float*)d_in[9];
  float* out = (float*)d_out;

  float*  nrm     = (float*)d_ws;                                  // 64 KB
  double* partial = (double*)((char*)d_ws + 2 * NROW * sizeof(float)); // 96 KB

  functor_kernel<<<dim3(64, 2, 1), 256, 0, stream>>>(
      XA, XB, WA1, bA1, WA2, bA2, WB1, bB1, WB2, bB2, out);
  norms_kernel<<<dim3(64, 1, 1), 256, 0, stream>>>(out, nrm);
  mmd_tile_kernel<<<dim3(64, 64, 3), 256, 0, stream>>>(out, nrm, partial);
  mmd_final_kernel<<<1, 256, 0, stream>>>(partial, out);
}